// myBlur_22187801051677
// MI455X (gfx1250) — compile-verified
//
#include <hip/hip_runtime.h>
#include <math.h>

// Separable 121-tap Gaussian blur as two banded GEMMs on the fp32 WMMA pipe.
// Per 16x16 tile: D = A(16x136) x B(136x16), K chunked into 34 chained
// v_wmma_f32_16x16x4_f32. Band matrix lives in a zero-padded LDS table
// (no branches); image slabs are staged into LDS with coalesced loads.

#define KSZ    121
#define GMEAN  60
#define HW     512
#define PLANES 24            // 8 batch * 3 channels
#define NSTEP  34            // ceil(136/4)
#define WPB    4             // waves (tiles) per 128-thread block
#define NTILES (PLANES * 32 * 32)
#define TSTR   138           // LDS slab stride: even (b64 align), 138%64=10 (no conflicts)
#define GPAD   160           // padded gaussian table: index = tap + 16, taps -16..143

typedef __attribute__((ext_vector_type(2))) float v2f;
typedef __attribute__((ext_vector_type(8))) float v8f;

__device__ __forceinline__ int clampi(int v, int lo, int hi) {
    return v < lo ? lo : (v > hi ? hi : v);
}

// ---- normalized 1D gaussian (2D normalization factorizes exactly) ----
__global__ void gauss1d_22187801051677(const float* __restrict__ sigma,
                                       float* __restrict__ g) {
    __shared__ float u[KSZ];
    int i = threadIdx.x;
    float s = sigma[0] * 8.0f + 16.0f;
    float inv2var = 1.0f / (2.0f * s * s);
    if (i < KSZ) {
        float d = (float)(i - GMEAN);
        u[i] = expf(-d * d * inv2var);
    }
    __syncthreads();
    if (i == 0) {
        float sum = 0.0f;
        for (int k = 0; k < KSZ; ++k) sum += u[k];
        float inv = 1.0f / sum;
        for (int k = 0; k < KSZ; ++k) g[k] = u[k] * inv;
    }
}

// ---- horizontal pass: D = In(16x136) x Band(136x16), Band[k][n] = g[k-n] ----
__global__ void __launch_bounds__(WPB * 32)
hpass_22187801051677(const float* __restrict__ in, const float* __restrict__ g,
                     float* __restrict__ h) {
    __shared__ float gp[GPAD];
    __shared__ float slab[WPB][16 * TSTR];

    const int wave = threadIdx.x >> 5;
    const int lane = threadIdx.x & 31;
    const int tile = blockIdx.x * WPB + wave;
    const int plane = tile >> 10;
    const int y0 = ((tile >> 5) & 31) << 4;
    const int x0 = (tile & 31) << 4;

    // zero-padded band table: gp[16 + t] = g[t] for t in [0,120], else 0
    for (int t = threadIdx.x; t < GPAD; t += blockDim.x) {
        int q = t - 16;
        gp[t] = (q >= 0 && q < KSZ) ? g[q] : 0.0f;
    }
    // stage 16 rows x 136 cols (replicate-pad via column clamp), coalesced
    const float* __restrict__ base = in + (size_t)plane * (HW * HW);
    for (int r = 0; r < 16; ++r) {
        const float* __restrict__ grow = base + (size_t)(y0 + r) * HW;
        for (int c = lane; c < 136; c += 32)
            slab[wave][r * TSTR + c] = grow[clampi(x0 - GMEAN + c, 0, HW - 1)];
    }
    __syncthreads();

    const int mn   = lane & 15;          // A-row (M) == B-col (N)
    const int koff = (lane >> 4) << 1;   // lanes 16-31 hold K+2

    const float* __restrict__ aRow = &slab[wave][mn * TSTR + koff];
    const float* __restrict__ bRow = &gp[16 + koff - mn];

    v8f acc = {};
#pragma unroll
    for (int j = 0; j < NSTEP; ++j) {
        v2f a; a.x = aRow[4 * j]; a.y = aRow[4 * j + 1];
        v2f b; b.x = bRow[4 * j]; b.y = bRow[4 * j + 1];
        acc = __builtin_amdgcn_wmma_f32_16x16x4_f32(false, a, false, b,
                                                    (short)0, acc, false, false);
    }

    float* __restrict__ outp = h + (size_t)plane * (HW * HW);
    const int rb = (lane >> 4) << 3;
#pragma unroll
    for (int r = 0; r < 8; ++r)
        outp[(size_t)(y0 + rb + r) * HW + x0 + mn] = acc[r];
}

// ---- vertical pass: D = Band(16x136) x H(136x16), Band[m][k] = g[k-m] ----
__global__ void __launch_bounds__(WPB * 32)
vpass_22187801051677(const float* __restrict__ h, const float* __restrict__ g,
                     float* __restrict__ out) {
    __shared__ float gp[GPAD];
    __shared__ float slabT[WPB][16 * TSTR];   // transposed: [col][row]

    const int wave = threadIdx.x >> 5;
    const int lane = threadIdx.x & 31;
    const int tile = blockIdx.x * WPB + wave;
    const int plane = tile >> 10;
    const int y0 = ((tile >> 5) & 31) << 4;
    const int x0 = (tile & 31) << 4;

    for (int t = threadIdx.x; t < GPAD; t += blockDim.x) {
        int q = t - 16;
        gp[t] = (q >= 0 && q < KSZ) ? g[q] : 0.0f;
    }
    // stage 136 rows x 16 cols (replicate-pad via row clamp), coalesced reads,
    // transposed store so K runs contiguously per lane
    const float* __restrict__ src = h + (size_t)plane * (HW * HW);
    for (int i = 0; i < 68; ++i) {            // 136*16/32 == 68 exactly
        int idx = (i << 5) + lane;
        int row = idx >> 4;
        int col = idx & 15;
        int gr = clampi(y0 - GMEAN + row, 0, HW - 1);
        slabT[wave][col * TSTR + row] = src[(size_t)gr * HW + x0 + col];
    }
    __syncthreads();

    const int mn   = lane & 15;
    const int koff = (lane >> 4) << 1;

    const float* __restrict__ aRow = &gp[16 + koff - mn];           // band
    const float* __restrict__ bRow = &slabT[wave][mn * TSTR + koff]; // data

    v8f acc = {};
#pragma unroll
    for (int j = 0; j < NSTEP; ++j) {
        v2f a; a.x = aRow[4 * j]; a.y = aRow[4 * j + 1];
        v2f b; b.x = bRow[4 * j]; b.y = bRow[4 * j + 1];
        acc = __builtin_amdgcn_wmma_f32_16x16x4_f32(false, a, false, b,
                                                    (short)0, acc, false, false);
    }

    float* __restrict__ outp = out + (size_t)plane * (HW * HW);
    const int rb = (lane >> 4) << 3;
#pragma unroll
    for (int r = 0; r < 8; ++r)
        outp[(size_t)(y0 + rb + r) * HW + x0 + mn] = acc[r];
}

extern "C" void kernel_launch(void* const* d_in, const int* in_sizes, int n_in,
                              void* d_out, int out_size, void* d_ws, size_t ws_size,
                              hipStream_t stream) {
    const float* x     = (const float*)d_in[0];   // [8,3,512,512] f32
    const float* sigma = (const float*)d_in[1];   // [1] f32
    float* outp = (float*)d_out;                  // [8,3,512,512] f32

    float* gbuf = (float*)d_ws;                   // 121 floats (padded region)
    float* hbuf = gbuf + 128;                     // 24*512*512 floats intermediate

    gauss1d_22187801051677<<<1, 128, 0, stream>>>(sigma, gbuf);
    hpass_22187801051677<<<NTILES / WPB, WPB * 32, 0, stream>>>(x, gbuf, hbuf);
    vpass_22187801051677<<<NTILES / WPB, WPB * 32, 0, stream>>>(hbuf, gbuf, outp);
}